// channel_attention_81655918232161
// MI455X (gfx1250) — compile-verified
//
#include <hip/hip_runtime.h>
#include <math.h>

typedef __attribute__((ext_vector_type(2))) float v2f;
typedef __attribute__((ext_vector_type(8))) float v8f;

#define CDIM   128
#define SDIM   400
#define SCH    80            // s-chunk (multiple of 10 and 16)
#define NCHUNK 5
#define PSTR   129           // padded stride for [*,128] LDS arrays
#define QPSTR  41            // padded stride for pooled [128,40] arrays
#define EPSLN  1e-5f

// ---- LDS layout (floats), total 42064 floats = 168256 bytes ----
#define XT_OFF   0                      // 80*129 = 10320
#define QB_OFF   (XT_OFF + SCH*PSTR)    // 10320
#define KB_OFF   (QB_OFF + SCH*PSTR)    // 10320
#define QP_OFF   (KB_OFF + SCH*PSTR)    // 128*41 = 5248
#define KP_OFF   (QP_OFF + CDIM*QPSTR)  // 5248
#define XMU_OFF  (KP_OFF + CDIM*QPSTR)  // 80
#define XRS_OFF  (XMU_OFF + SCH)
#define QMU_OFF  (XRS_OFF + SCH)
#define QRS_OFF  (QMU_OFF + SCH)
#define KMU_OFF  (QRS_OFF + SCH)
#define KRS_OFF  (KMU_OFF + SCH)
#define RSUM_OFF (KRS_OFF + SCH)        // 128
#define SMEM_FLOATS (RSUM_OFF + CDIM)
// attention matrix [128,129] aliased over qb+kb (16512 <= 20640)
#define ATT_OFF  QB_OFF

__device__ __forceinline__ v8f wmma4(v2f a, v2f b, v8f c) {
  return __builtin_amdgcn_wmma_f32_16x16x4_f32(
      /*neg_a=*/false, a, /*neg_b=*/false, b,
      /*c_mod=*/(short)0, c, /*reuse_a=*/false, /*reuse_b=*/false);
}

// Stage x[c, s0:s0+80] transposed into LDS xt[s][c] (stride PSTR), float4
// loads, optional per-channel scale (used in pass 2 to fuse *rowsum[c]).
__device__ __forceinline__ void load_xt(const float* __restrict__ xb, int s0,
                                        int tid, float* __restrict__ xt,
                                        float scale) {
  int c = tid >> 2, qd = tid & 3;
#pragma unroll
  for (int i = 0; i < 5; ++i) {
    int sl = qd * 20 + i * 4;
    float4 v = *(const float4*)(xb + c * SDIM + s0 + sl);
    xt[(sl + 0) * PSTR + c] = v.x * scale;
    xt[(sl + 1) * PSTR + c] = v.y * scale;
    xt[(sl + 2) * PSTR + c] = v.z * scale;
    xt[(sl + 3) * PSTR + c] = v.w * scale;
  }
}

// Per-row mean / rstd over 128 cols for 80 rows (4 lanes per row).
__device__ __forceinline__ void row_stats(const float* __restrict__ buf, int tid,
                                          float* __restrict__ mus,
                                          float* __restrict__ rss) {
  if (tid < 4 * SCH) {
    int row = tid >> 2, part = tid & 3;
    const float* p = buf + row * PSTR + part * 32;
    float s = 0.f, s2 = 0.f;
#pragma unroll
    for (int j = 0; j < 32; ++j) { float v = p[j]; s += v; s2 += v * v; }
    s  += __shfl_xor(s, 1, 4);  s  += __shfl_xor(s, 2, 4);
    s2 += __shfl_xor(s2, 1, 4); s2 += __shfl_xor(s2, 2, 4);
    float mu = s * (1.0f / 128.0f);
    float var = s2 * (1.0f / 128.0f) - mu * mu;
    if (part == 0) { mus[row] = mu; rss[row] = rsqrtf(var + EPSLN); }
  }
}

__launch_bounds__(512, 1)
__global__ void channel_attention_fused(
    const float* __restrict__ x,
    const float* __restrict__ Wq, const float* __restrict__ bq,
    const float* __restrict__ gq, const float* __restrict__ betaq,
    const float* __restrict__ Wk, const float* __restrict__ bk,
    const float* __restrict__ gk, const float* __restrict__ betak,
    const float* __restrict__ Wp, const float* __restrict__ bp,
    const float* __restrict__ gp, const float* __restrict__ betap,
    float* __restrict__ out) {
  extern __shared__ float smem[];
  float* xt  = smem + XT_OFF;
  float* qb  = smem + QB_OFF;
  float* kb  = smem + KB_OFF;
  float* qp  = smem + QP_OFF;
  float* kp  = smem + KP_OFF;
  float* att = smem + ATT_OFF;
  float* xmu = smem + XMU_OFF; float* xrs = smem + XRS_OFF;
  float* qmu = smem + QMU_OFF; float* qrs = smem + QRS_OFF;
  float* kmu = smem + KMU_OFF; float* krs = smem + KRS_OFF;
  float* rowsum = smem + RSUM_OFF;

  const int tid  = threadIdx.x;
  const int wave = tid >> 5;
  const int lane = tid & 31;
  const int bo   = blockIdx.x;
  const float* xb = x + (size_t)bo * CDIM * SDIM;
  float* ob       = out + (size_t)bo * CDIM * SDIM;

  const int ntile = wave & 7;
  const int nidx  = ntile * 16 + (lane & 15);  // output-channel this lane owns
  const int khalf = (lane >> 4) * 2;           // K offset within 4-wide step
  const int arow  = lane & 15;                 // A-fragment row within tile

  // ---------------- Pass 1: q/k GEMM weight fragments (persistent in VGPRs)
  const float* Wqk = (wave < 8) ? Wq : Wk;
  v2f bf[32];
#pragma unroll
  for (int ks = 0; ks < 32; ++ks) {
    int k0 = ks * 4 + khalf;
    bf[ks].x = Wqk[nidx * CDIM + k0];
    bf[ks].y = Wqk[nidx * CDIM + k0 + 1];
  }
  const float biasqk = (wave < 8) ? bq[nidx] : bk[nidx];
  float* dst = (wave < 8) ? qb : kb;

  for (int ch = 0; ch < NCHUNK; ++ch) {
    const int s0 = ch * SCH;
    load_xt(xb, s0, tid, xt, 1.0f);
    __syncthreads();
    row_stats(xt, tid, xmu, xrs);   // LN stats of x^T rows
    __syncthreads();

    // Normalize xt in place ONCE (shared by all 16 waves' A-fragments).
#pragma unroll
    for (int j = 0; j < 20; ++j) {
      int e = tid + j * 512;
      int row = e >> 7, col = e & 127;
      xt[row * PSTR + col] = (xt[row * PSTR + col] - xmu[row]) * xrs[row];
    }
    __syncthreads();

    // q / k GEMM: each wave covers one 16-wide n-stripe, all 5 m-tiles.
    // Inner loop is now just ds_load_2addr + v_wmma.
#pragma unroll 1
    for (int mt = 0; mt < 5; ++mt) {
      const int row = mt * 16 + arow;
      const float* arowp = xt + row * PSTR;
      v8f acc = {};
#pragma unroll
      for (int ks = 0; ks < 32; ++ks) {
        int k0 = ks * 4 + khalf;
        v2f a;
        a.x = arowp[k0];
        a.y = arowp[k0 + 1];
        acc = wmma4(a, bf[ks], acc);
      }
      const int rb = mt * 16 + (lane >> 4) * 8;
#pragma unroll
      for (int v = 0; v < 8; ++v)
        dst[(rb + v) * PSTR + nidx] = acc[v] + biasqk;
    }
    __syncthreads();

    row_stats(qb, tid, qmu, qrs);
    row_stats(kb, tid, kmu, krs);
    __syncthreads();

    // LN + AvgPool(10) fused: pooled = g*(mean of normalized) + beta
#pragma unroll
    for (int it = 0; it < 2; ++it) {
      int task = tid + it * 512;
      int c = task & 127, t8 = task >> 7;
      float sq = 0.f, sk = 0.f;
#pragma unroll
      for (int i = 0; i < 10; ++i) {
        int sl = t8 * 10 + i;
        sq += (qb[sl * PSTR + c] - qmu[sl]) * qrs[sl];
        sk += (kb[sl * PSTR + c] - kmu[sl]) * krs[sl];
      }
      int tg = ch * 8 + t8;
      qp[c * QPSTR + tg] = gq[c] * (sq * 0.1f) + betaq[c];
      kp[c * QPSTR + tg] = gk[c] * (sk * 0.1f) + betak[c];
    }
    __syncthreads();
  }

  // ---------------- Attention: atten = qp * kp^T / sqrt(40), 8x8 tiles
  const float invsc = 0.15811388300841897f;  // 1/sqrt(40)
#pragma unroll 1
  for (int j = 0; j < 4; ++j) {
    const int tileid = wave * 4 + j;
    const int mt = tileid >> 3, nt = tileid & 7;
    const int am = mt * 16 + (lane & 15);
    const int bn = nt * 16 + (lane & 15);
    v8f acc = {};
#pragma unroll
    for (int ks = 0; ks < 10; ++ks) {
      int k0 = ks * 4 + khalf;
      v2f a, b;
      a.x = qp[am * QPSTR + k0]; a.y = qp[am * QPSTR + k0 + 1];
      b.x = kp[bn * QPSTR + k0]; b.y = kp[bn * QPSTR + k0 + 1];
      acc = wmma4(a, b, acc);
    }
    const int rb = mt * 16 + (lane >> 4) * 8;
    const int cb = nt * 16 + (lane & 15);
#pragma unroll
    for (int v = 0; v < 8; ++v)
      att[(rb + v) * PSTR + cb] = acc[v] * invsc;
  }
  __syncthreads();

  // ---------------- Softmax row-sum (4 lanes per row, 128 rows)
  {
    int row = tid >> 2, part = tid & 3;
    float* p = att + row * PSTR + part * 32;
    float mx = -INFINITY;
#pragma unroll
    for (int jj = 0; jj < 32; ++jj) mx = fmaxf(mx, p[jj]);
    mx = fmaxf(mx, __shfl_xor(mx, 1, 4));
    mx = fmaxf(mx, __shfl_xor(mx, 2, 4));
    float se = 0.f;
#pragma unroll
    for (int jj = 0; jj < 32; ++jj) { float e = __expf(p[jj] - mx); p[jj] = e; se += e; }
    se += __shfl_xor(se, 1, 4); se += __shfl_xor(se, 2, 4);
    float rz = 1.0f / se;
    float ps = 0.f;
#pragma unroll
    for (int jj = 0; jj < 32; ++jj) ps += p[jj] * rz;
    ps += __shfl_xor(ps, 1, 4); ps += __shfl_xor(ps, 2, 4);
    if (part == 0) rowsum[row] = ps;
  }
  __syncthreads();

  // ---------------- Pass 2: p-GEMM weight fragments (reuse bf registers)
#pragma unroll
  for (int ks = 0; ks < 32; ++ks) {
    int k0 = ks * 4 + khalf;
    bf[ks].x = Wp[nidx * CDIM + k0];
    bf[ks].y = Wp[nidx * CDIM + k0 + 1];
  }
  const float biasp = bp[nidx];

  for (int ch = 0; ch < NCHUNK; ++ch) {
    const int s0 = ch * SCH;
    // Fuse the per-channel *rowsum[c] into the staging pass: each loader
    // thread owns a single channel column c = tid>>2.
    load_xt(xb, s0, tid, xt, rowsum[tid >> 2]);
    __syncthreads();

    // y = (x^T scaled-by-rowsum) @ Wp^T + bp ; 40 tiles over 16 waves
#pragma unroll 1
    for (int id = wave; id < 40; id += 16) {
      const int mt = id >> 3;          // nt == wave&7 for every id (16 % 8 == 0)
      const int row = mt * 16 + arow;
      const float* arowp = xt + row * PSTR;
      v8f acc = {};
#pragma unroll
      for (int ks = 0; ks < 32; ++ks) {
        int k0 = ks * 4 + khalf;
        v2f a;
        a.x = arowp[k0];
        a.y = arowp[k0 + 1];
        acc = wmma4(a, bf[ks], acc);
      }
      const int rb = mt * 16 + (lane >> 4) * 8;
#pragma unroll
      for (int v = 0; v < 8; ++v)
        qb[(rb + v) * PSTR + nidx] = acc[v] + biasp;
    }
    __syncthreads();

    row_stats(qb, tid, qmu, qrs);
    __syncthreads();

    // out[c][s] = LN(y)[s][c]*gp[c] + betap[c] + x[c][s]
    {
      int c = tid >> 2, qd = tid & 3;
      const float g = gp[c], be = betap[c];
#pragma unroll
      for (int i = 0; i < 5; ++i) {
        int sl = qd * 20 + i * 4;
        float4 xv = *(const float4*)(xb + c * SDIM + s0 + sl);
        float4 ov;
        ov.x = (qb[(sl + 0) * PSTR + c] - qmu[sl + 0]) * qrs[sl + 0] * g + be + xv.x;
        ov.y = (qb[(sl + 1) * PSTR + c] - qmu[sl + 1]) * qrs[sl + 1] * g + be + xv.y;
        ov.z = (qb[(sl + 2) * PSTR + c] - qmu[sl + 2]) * qrs[sl + 2] * g + be + xv.z;
        ov.w = (qb[(sl + 3) * PSTR + c] - qmu[sl + 3]) * qrs[sl + 3] * g + be + xv.w;
        *(float4*)(ob + c * SDIM + s0 + sl) = ov;
      }
    }
    __syncthreads();
  }
}

extern "C" void kernel_launch(void* const* d_in, const int* in_sizes, int n_in,
                              void* d_out, int out_size, void* d_ws, size_t ws_size,
                              hipStream_t stream) {
  (void)in_sizes; (void)n_in; (void)out_size; (void)d_ws; (void)ws_size;
  const float* x     = (const float*)d_in[0];
  const float* Wq    = (const float*)d_in[1];
  const float* bq    = (const float*)d_in[2];
  const float* gq    = (const float*)d_in[3];
  const float* betaq = (const float*)d_in[4];
  const float* Wk    = (const float*)d_in[5];
  const float* bk    = (const float*)d_in[6];
  const float* gk    = (const float*)d_in[7];
  const float* betak = (const float*)d_in[8];
  const float* Wp    = (const float*)d_in[9];
  const float* bp    = (const float*)d_in[10];
  const float* gp    = (const float*)d_in[11];
  const float* betap = (const float*)d_in[12];
  float* outp        = (float*)d_out;

  const size_t smem_bytes = (size_t)SMEM_FLOATS * sizeof(float);  // 168256 B
  // Allow >64KB dynamic LDS (CDNA5 WGP has 320KB). Idempotent host-side call.
  (void)hipFuncSetAttribute((const void*)channel_attention_fused,
                            hipFuncAttributeMaxDynamicSharedMemorySize,
                            (int)smem_bytes);

  dim3 grid(64 * 16);   // one workgroup per (b, o) slice
  dim3 block(512);      // 16 wave32
  channel_attention_fused<<<grid, block, smem_bytes, stream>>>(
      x, Wq, bq, gq, betaq, Wk, bk, gk, betak, Wp, bp, gp, betap, outp);
}